// TwoLayerFSL_19095424598299
// MI455X (gfx1250) — compile-verified
//
#include <hip/hip_runtime.h>
#include <math.h>

typedef __attribute__((ext_vector_type(2))) float v2f;
typedef __attribute__((ext_vector_type(8))) float v8f;

// ---------------------------------------------------------------------------
// Kernel 1: degree accumulation  deg[dst] += 1
// ---------------------------------------------------------------------------
__global__ void deg_kernel(const int* __restrict__ ei, float* __restrict__ deg, int E) {
    int e = blockIdx.x * blockDim.x + threadIdx.x;
    if (e < E) atomicAdd(&deg[ei[E + e]], 1.0f);
}

// ---------------------------------------------------------------------------
// Kernel 2: in-place deg -> deg^{-1/2}  (0 if deg == 0)
// ---------------------------------------------------------------------------
__global__ void isqrt_kernel(float* __restrict__ deg, int N) {
    int i = blockIdx.x * blockDim.x + threadIdx.x;
    if (i < N) {
        float d = deg[i];
        deg[i] = (d > 0.0f) ? rsqrtf(d) : 0.0f;
    }
}

// ---------------------------------------------------------------------------
// Kernel 3: h1 = x @ W1 + b1   (N x 512) @ (512 x 32), f32 WMMA 16x16x4
// Block: 128 threads = 4 waves; block tile = 32 rows x 32 cols.
// LDS row stride 36 floats: 144 B = 16B-aligned for b128 stores AND
// conflict-free for 16-lane column reads (36*l mod 64 distinct, l=0..15).
// ---------------------------------------------------------------------------
__global__ __launch_bounds__(128) void gemm1_kernel(
    const float* __restrict__ x, const float* __restrict__ W1,
    const float* __restrict__ b1, float* __restrict__ h1, int N, int K)
{
    __shared__ __align__(16) float As[32][36];  // 32 rows x 32 k
    __shared__ __align__(16) float Bs[32][36];  // 32 k x 32 cols

    const int tid  = threadIdx.x;
    const int wave = tid >> 5;
    const int lane = tid & 31;
    const int row0 = blockIdx.x * 32;
    const int wm   = (wave >> 1) << 4;
    const int wn   = (wave & 1) << 4;
    const int half = lane >> 4;   // 0: K pair {0,1}, 1: K pair {2,3}
    const int l    = lane & 15;

    v8f acc = {};

    for (int k0 = 0; k0 < K; k0 += 32) {
        // cooperative load: 2 tiles x 256 float4 with 128 threads (2 each)
        #pragma unroll
        for (int i = 0; i < 2; ++i) {
            int linear = i * 128 + tid;          // 0..255
            int r = linear >> 3, c4 = linear & 7;
            int gr = row0 + r;
            float4 av = make_float4(0.f, 0.f, 0.f, 0.f);
            if (gr < N)
                av = *(const float4*)(x + (long long)gr * K + (k0 + c4 * 4));
            *(float4*)&As[r][c4 * 4] = av;
            *(float4*)&Bs[r][c4 * 4] =
                *(const float4*)(W1 + (long long)(k0 + r) * 32 + c4 * 4);
        }
        __syncthreads();

        #pragma unroll
        for (int kk = 0; kk < 32; kk += 4) {
            v2f a, b;
            // A 16x4: lanes 0-15 hold K=kk,kk+1 ; lanes 16-31 hold K=kk+2,kk+3
            a.x = As[wm + l][kk + 2 * half + 0];
            a.y = As[wm + l][kk + 2 * half + 1];
            // B 4x16 mirrors A: lane n holds column n for the same K pair
            b.x = Bs[kk + 2 * half + 0][wn + l];
            b.y = Bs[kk + 2 * half + 1][wn + l];
            acc = __builtin_amdgcn_wmma_f32_16x16x4_f32(
                false, a, false, b, (short)0, acc, false, false);
        }
        __syncthreads();
    }

    // C/D layout: VGPR r -> M = r (lanes 0-15) / M = r+8 (lanes 16-31), N = lane&15
    const int rofs = half * 8;
    #pragma unroll
    for (int r = 0; r < 8; ++r) {
        int row = row0 + wm + rofs + r;
        int col = wn + l;
        if (row < N) h1[(long long)row * 32 + col] = acc[r] + b1[col];
    }
}

// ---------------------------------------------------------------------------
// Kernel 4: layer-1 scatter. One wave (32 lanes) per edge, lane = feature.
// agg1[dst, f] += h1[src, f] * dinv[src] * dinv[dst]
// ---------------------------------------------------------------------------
__global__ void scatter1_kernel(const float* __restrict__ h1,
                                const int* __restrict__ ei,
                                const float* __restrict__ dinv,
                                float* __restrict__ agg1, int E)
{
    int wid  = (blockIdx.x * blockDim.x + threadIdx.x) >> 5;
    int lane = threadIdx.x & 31;
    if (wid >= E) return;
    int s = ei[wid];
    int d = ei[E + wid];
    float nrm = dinv[s] * dinv[d];
    float v = h1[s * 32 + lane] * nrm;
    atomicAdd(&agg1[d * 32 + lane], v);
}

// ---------------------------------------------------------------------------
// Kernel 5: h2 = ELU(agg1) @ W2 + b2   (N x 32) @ (32 x 40), cols padded to 48.
// Block: 96 threads = 3 waves; block tile = 16 rows x 48 cols; K = 32.
// ---------------------------------------------------------------------------
__global__ __launch_bounds__(96) void gemm2_kernel(
    const float* __restrict__ agg1, const float* __restrict__ W2,
    const float* __restrict__ b2, float* __restrict__ h2, int N, int FOUT)
{
    __shared__ __align__(16) float As[16][36];  // 16 rows x 32 k
    __shared__ __align__(16) float Bs[32][48];  // 32 k x 48 cols (padded)

    const int tid  = threadIdx.x;
    const int wave = tid / 32;
    const int lane = tid & 31;
    const int row0 = blockIdx.x * 16;
    const int wn   = wave * 16;
    const int half = lane >> 4;
    const int l    = lane & 15;

    // Load A as float4 with fused ELU (agg1 rows are 128B, 16B-aligned)
    for (int linear = tid; linear < 16 * 8; linear += 96) {
        int r = linear >> 3, c4 = linear & 7;
        int gr = row0 + r;
        float4 v = make_float4(0.f, 0.f, 0.f, 0.f);
        if (gr < N) v = *(const float4*)(agg1 + gr * 32 + c4 * 4);
        v.x = (v.x > 0.0f) ? v.x : (expf(v.x) - 1.0f);
        v.y = (v.y > 0.0f) ? v.y : (expf(v.y) - 1.0f);
        v.z = (v.z > 0.0f) ? v.z : (expf(v.z) - 1.0f);
        v.w = (v.w > 0.0f) ? v.w : (expf(v.w) - 1.0f);
        *(float4*)&As[r][c4 * 4] = v;
    }
    // Load B scalar (40 not float4-divisible; only 1280 elems), zero-pad cols
    for (int linear = tid; linear < 32 * 48; linear += 96) {
        int r = linear / 48, c = linear % 48;
        Bs[r][c] = (c < FOUT) ? W2[r * FOUT + c] : 0.0f;
    }
    __syncthreads();

    v8f acc = {};
    #pragma unroll
    for (int kk = 0; kk < 32; kk += 4) {
        v2f a, b;
        a.x = As[l][kk + 2 * half + 0];
        a.y = As[l][kk + 2 * half + 1];
        b.x = Bs[kk + 2 * half + 0][wn + l];
        b.y = Bs[kk + 2 * half + 1][wn + l];
        acc = __builtin_amdgcn_wmma_f32_16x16x4_f32(
            false, a, false, b, (short)0, acc, false, false);
    }

    const int rofs = half * 8;
    #pragma unroll
    for (int r = 0; r < 8; ++r) {
        int row = row0 + rofs + r;
        int col = wn + l;
        if (row < N && col < FOUT) h2[(long long)row * FOUT + col] = acc[r] + b2[col];
    }
}

// ---------------------------------------------------------------------------
// Kernel 6: layer-2 scatter. Thread per (edge, feature).
// out[dst, f] += h2[src, f] * dinv[src] * dinv[dst]
// ---------------------------------------------------------------------------
__global__ void scatter2_kernel(const float* __restrict__ h2,
                                const int* __restrict__ ei,
                                const float* __restrict__ dinv,
                                float* __restrict__ out, int E, int FOUT)
{
    long long tid = (long long)blockIdx.x * blockDim.x + threadIdx.x;
    long long total = (long long)E * FOUT;
    if (tid >= total) return;
    int e = (int)(tid / FOUT);
    int f = (int)(tid % FOUT);
    int s = ei[e];
    int d = ei[E + e];
    float nrm = dinv[s] * dinv[d];
    atomicAdd(&out[(long long)d * FOUT + f], h2[(long long)s * FOUT + f] * nrm);
}

// ---------------------------------------------------------------------------
// Kernel 7: in-place log_softmax over rows of 40. One wave per row.
// Lane covers f = lane and (lane < 8) f = 32 + lane.
// ---------------------------------------------------------------------------
__global__ void logsoftmax_kernel(float* __restrict__ out, int N, int FOUT)
{
    int row  = (blockIdx.x * blockDim.x + threadIdx.x) >> 5;
    int lane = threadIdx.x & 31;
    if (row >= N) return;
    float* p = out + (long long)row * FOUT;
    float v0 = p[lane];
    float v1 = (lane < FOUT - 32) ? p[32 + lane] : -3.4e38f;
    float m = fmaxf(v0, v1);
    #pragma unroll
    for (int off = 16; off > 0; off >>= 1)
        m = fmaxf(m, __shfl_xor(m, off, 32));
    float s = expf(v0 - m) + ((lane < FOUT - 32) ? expf(v1 - m) : 0.0f);
    #pragma unroll
    for (int off = 16; off > 0; off >>= 1)
        s += __shfl_xor(s, off, 32);
    float ls = m + logf(s);
    p[lane] = v0 - ls;
    if (lane < FOUT - 32) p[32 + lane] = v1 - ls;
}

// ---------------------------------------------------------------------------
static inline size_t alignUp(size_t v) { return (v + 255) & ~(size_t)255; }

extern "C" void kernel_launch(void* const* d_in, const int* in_sizes, int n_in,
                              void* d_out, int out_size, void* d_ws, size_t ws_size,
                              hipStream_t stream) {
    const float* x  = (const float*)d_in[0];
    const int*   ei = (const int*)d_in[1];   // edge_index [2, E]
    const float* W1 = (const float*)d_in[2];
    const float* b1 = (const float*)d_in[3];
    const float* W2 = (const float*)d_in[4];
    const float* b2 = (const float*)d_in[5];
    float* out = (float*)d_out;

    const int HID  = in_sizes[3];              // 32
    const int FOUT = in_sizes[5];              // 40
    const int FIN  = in_sizes[2] / HID;        // 512
    const int N    = in_sizes[0] / FIN;        // 50000
    const int E    = in_sizes[1] / 2;          // 1600000

    // Workspace layout
    char* ws = (char*)d_ws;
    float* dinv = (float*)ws; ws += alignUp((size_t)N * 4);
    float* h1   = (float*)ws; ws += alignUp((size_t)N * HID * 4);
    float* agg1 = (float*)ws; ws += alignUp((size_t)N * HID * 4);
    float* h2   = (float*)ws;

    // Zero atomic-accumulation targets (deterministic each call)
    hipMemsetAsync(dinv, 0, (size_t)N * 4, stream);
    hipMemsetAsync(agg1, 0, (size_t)N * HID * 4, stream);
    hipMemsetAsync(out,  0, (size_t)N * FOUT * 4, stream);

    // 1. degrees
    deg_kernel<<<(E + 255) / 256, 256, 0, stream>>>(ei, dinv, E);
    // 2. deg^{-1/2}
    isqrt_kernel<<<(N + 255) / 256, 256, 0, stream>>>(dinv, N);
    // 3. h1 = x @ W1 + b1 (WMMA)
    gemm1_kernel<<<(N + 31) / 32, 128, 0, stream>>>(x, W1, b1, h1, N, FIN);
    // 4. scatter layer 1 (wave per edge)
    {
        long long threads = (long long)E * 32;
        scatter1_kernel<<<(unsigned)((threads + 255) / 256), 256, 0, stream>>>(
            h1, ei, dinv, agg1, E);
    }
    // 5. h2 = ELU(agg1) @ W2 + b2 (WMMA, fused ELU)
    gemm2_kernel<<<(N + 15) / 16, 96, 0, stream>>>(agg1, W2, b2, h2, N, FOUT);
    // 6. scatter layer 2 (thread per edge-feature)
    {
        long long threads = (long long)E * FOUT;
        scatter2_kernel<<<(unsigned)((threads + 255) / 256), 256, 0, stream>>>(
            h2, ei, dinv, out, E, FOUT);
    }
    // 7. log_softmax rows (wave per row)
    {
        long long threads = (long long)N * 32;
        logsoftmax_kernel<<<(unsigned)((threads + 255) / 256), 256, 0, stream>>>(
            out, N, FOUT);
    }
}